// HeteroGraphSAGE_12790412607511
// MI455X (gfx1250) — compile-verified
//
#include <hip/hip_runtime.h>

// ---------------------------------------------------------------------------
// HeteroGraphSAGE on MI455X (gfx1250, wave32)
//  N=100000 nodes/type, E=500000 edges/rel, D=H=128, fp32.
//  - seg-mean via wave-per-edge float4 atomics (L2-resident, 192MB L2)
//  - dense layers via V_WMMA_F32_16X16X4_F32 (full fp32 matrix path)
// ---------------------------------------------------------------------------

typedef __attribute__((ext_vector_type(2))) float v2f;
typedef __attribute__((ext_vector_type(8))) float v8f;

#define GN 100000
#define GE 500000
#define GD 128

__device__ __forceinline__ void atomAddF(float* p, float v) {
  __hip_atomic_fetch_add(p, v, __ATOMIC_RELAXED, __HIP_MEMORY_SCOPE_AGENT);
}

// One wave32 per edge: lane L moves feat[src][4L..4L+3] -> atomic add into
// nsum[dst][4L..4L+3]. lane0 optionally bumps the in-degree count.
__global__ __launch_bounds__(256) void scatter_sum(
    const float* __restrict__ feat, const int* __restrict__ src,
    const int* __restrict__ dst, float* __restrict__ nsum,
    float* __restrict__ cnt, int nedges) {
  const int wid  = (int)((blockIdx.x * blockDim.x + threadIdx.x) >> 5);
  const int lane = threadIdx.x & 31;
  if (wid >= nedges) return;
  const int s = src[wid];
  const int d = dst[wid];
  const float4 v = *(const float4*)(feat + (size_t)s * GD + lane * 4);
  float* o = nsum + (size_t)d * GD + lane * 4;
  atomAddF(o + 0, v.x);
  atomAddF(o + 1, v.y);
  atomAddF(o + 2, v.z);
  atomAddF(o + 3, v.w);
  if (cnt != nullptr && lane == 0) atomAddF(cnt + d, 1.0f);
}

// Fused conv1: H = relu(X @ Ws^T + (NS/max(cnt,1)) @ Wn^T + bs + bn)
// One wave per 16x16 output tile, K=128 in steps of 4, 64 WMMAs per tile.
// f32 16x16x4 fragment layout (per ISA 7.12.2):
//   A (16x4): lane l<16 row=l holds K=k0,k0+1 ; lane>=16 same rows, K=k0+2,k0+3
//   B (4x16): lane l<16 col=l holds K=k0,k0+1 ; lane>=16 same cols, K=k0+2,k0+3
//   -> both fragments are contiguous per-lane float2 loads.
__global__ __launch_bounds__(128) void conv1_gemm(
    const float* __restrict__ X, const float* __restrict__ NS,
    const float* __restrict__ CNT, const float* __restrict__ Ws,
    const float* __restrict__ Wn, const float* __restrict__ bs,
    const float* __restrict__ bn, float* __restrict__ Hout) {
  const int lane = threadIdx.x & 31;
  const int wave = threadIdx.x >> 5;
  const int tile = (int)blockIdx.x * 4 + wave;   // (N/16)*8 tiles total
  const int tm   = tile >> 3;                    // row tile
  const int tn   = tile & 7;                     // col tile
  const int half = lane >> 4;
  const int l16  = lane & 15;

  const int arow = tm * 16 + l16;                // A row owned by this lane
  const int bcol = tn * 16 + l16;                // output column for this lane
  const float invc = 1.0f / fmaxf(CNT[arow], 1.0f);

  const float* __restrict__ xrow = X  + (size_t)arow * GD;
  const float* __restrict__ nrow = NS + (size_t)arow * GD;
  const float* __restrict__ wsr  = Ws + (size_t)bcol * GD;  // W row-major [out,in]
  const float* __restrict__ wnr  = Wn + (size_t)bcol * GD;

  v8f acc = {};
#pragma unroll
  for (int k0 = 0; k0 < GD; k0 += 4) {
    const int ka = k0 + 2 * half;
    v2f a_s = {xrow[ka], xrow[ka + 1]};
    v2f b_s = {wsr[ka],  wsr[ka + 1]};
    acc = __builtin_amdgcn_wmma_f32_16x16x4_f32(false, a_s, false, b_s,
                                                (short)0, acc, false, false);
    v2f a_n = {nrow[ka] * invc, nrow[ka + 1] * invc};
    v2f b_n = {wnr[ka], wnr[ka + 1]};
    acc = __builtin_amdgcn_wmma_f32_16x16x4_f32(false, a_n, false, b_n,
                                                (short)0, acc, false, false);
  }

  const float bias = bs[bcol] + bn[bcol];
#pragma unroll
  for (int v = 0; v < 8; ++v) {
    const int m = tm * 16 + v + half * 8;        // C/D layout: lanes>=16 -> M+8
    const float r = acc[v] + bias;
    Hout[(size_t)m * GD + bcol] = fmaxf(r, 0.0f);
  }
}

// conv2 (out_feats=1): one wave per node, float4 partial dots + xor-shuffle
// reduction across the 32 lanes.
__global__ __launch_bounds__(256) void out_row(
    const float* __restrict__ Hself, const float* __restrict__ NS,
    const float* __restrict__ CNT, const float* __restrict__ ws,
    const float* __restrict__ wn, const float* __restrict__ bsp,
    const float* __restrict__ bnp, float* __restrict__ out, int nrows) {
  const int wid  = (int)((blockIdx.x * blockDim.x + threadIdx.x) >> 5);
  const int lane = threadIdx.x & 31;
  if (wid >= nrows) return;
  const float invc = 1.0f / fmaxf(CNT[wid], 1.0f);
  const float4 h = *(const float4*)(Hself + (size_t)wid * GD + lane * 4);
  const float4 n = *(const float4*)(NS + (size_t)wid * GD + lane * 4);
  const float4 a = *(const float4*)(ws + lane * 4);
  const float4 b = *(const float4*)(wn + lane * 4);
  float p = h.x * a.x + h.y * a.y + h.z * a.z + h.w * a.w +
            invc * (n.x * b.x + n.y * b.y + n.z * b.z + n.w * b.w);
  p += __shfl_xor(p, 16);
  p += __shfl_xor(p, 8);
  p += __shfl_xor(p, 4);
  p += __shfl_xor(p, 2);
  p += __shfl_xor(p, 1);
  if (lane == 0) out[wid] = p + bsp[0] + bnp[0];
}

extern "C" void kernel_launch(void* const* d_in, const int* in_sizes, int n_in,
                              void* d_out, int out_size, void* d_ws,
                              size_t ws_size, hipStream_t stream) {
  (void)in_sizes; (void)n_in; (void)out_size; (void)ws_size;
  const float* x_user  = (const float*)d_in[0];
  const float* x_tweet = (const float*)d_in[1];
  const float* w1ps = (const float*)d_in[2];  const float* b1ps = (const float*)d_in[3];
  const float* w1pn = (const float*)d_in[4];  const float* b1pn = (const float*)d_in[5];
  const float* w1bs = (const float*)d_in[6];  const float* b1bs = (const float*)d_in[7];
  const float* w1bn = (const float*)d_in[8];  const float* b1bn = (const float*)d_in[9];
  const float* w2ps = (const float*)d_in[10]; const float* b2ps = (const float*)d_in[11];
  const float* w2pn = (const float*)d_in[12]; const float* b2pn = (const float*)d_in[13];
  const float* w2bs = (const float*)d_in[14]; const float* b2bs = (const float*)d_in[15];
  const float* w2bn = (const float*)d_in[16]; const float* b2bn = (const float*)d_in[17];
  const int* src_posts = (const int*)d_in[18];
  const int* dst_posts = (const int*)d_in[19];
  const int* src_pb    = (const int*)d_in[20];
  const int* dst_pb    = (const int*)d_in[21];

  // Workspace layout (floats): [ns_t][ns_u][cnt_t][cnt_u][h_t][h_u]
  float* ws_f  = (float*)d_ws;
  float* ns_t  = ws_f;
  float* ns_u  = ns_t + (size_t)GN * GD;
  float* cnt_t = ns_u + (size_t)GN * GD;
  float* cnt_u = cnt_t + GN;
  float* h_t   = cnt_u + GN;
  float* h_u   = h_t + (size_t)GN * GD;

  float* out_user  = (float*)d_out;       // reference returns (out_user, out_tweet)
  float* out_tweet = out_user + GN;

  const int scatterBlocks = (GE * 32) / 256;       // one wave per edge
  const int gemmBlocks    = (GN / 16) * 8 / 4;     // 4 tile-waves per block
  const int rowBlocks     = (GN * 32 + 255) / 256; // one wave per node

  // ---- layer 1 ----
  hipMemsetAsync(ns_t, 0, ((size_t)2 * GN * GD + 2 * GN) * sizeof(float), stream);
  scatter_sum<<<scatterBlocks, 256, 0, stream>>>(x_user, src_posts, dst_posts,
                                                 ns_t, cnt_t, GE);
  scatter_sum<<<scatterBlocks, 256, 0, stream>>>(x_tweet, src_pb, dst_pb,
                                                 ns_u, cnt_u, GE);
  conv1_gemm<<<gemmBlocks, 128, 0, stream>>>(x_tweet, ns_t, cnt_t,
                                             w1ps, w1pn, b1ps, b1pn, h_t);
  conv1_gemm<<<gemmBlocks, 128, 0, stream>>>(x_user, ns_u, cnt_u,
                                             w1bs, w1bn, b1bs, b1bn, h_u);

  // ---- layer 2 (counts unchanged; reuse sum buffers) ----
  hipMemsetAsync(ns_t, 0, (size_t)2 * GN * GD * sizeof(float), stream);
  scatter_sum<<<scatterBlocks, 256, 0, stream>>>(h_u, src_posts, dst_posts,
                                                 ns_t, nullptr, GE);
  scatter_sum<<<scatterBlocks, 256, 0, stream>>>(h_t, src_pb, dst_pb,
                                                 ns_u, nullptr, GE);
  out_row<<<rowBlocks, 256, 0, stream>>>(h_t, ns_t, cnt_t, w2ps, w2pn,
                                         b2ps, b2pn, out_tweet, GN);
  out_row<<<rowBlocks, 256, 0, stream>>>(h_u, ns_u, cnt_u, w2bs, w2bn,
                                         b2bs, b2bn, out_user, GN);
}